// HIVNet_4398046511479
// MI455X (gfx1250) — compile-verified
//
#include <hip/hip_runtime.h>
#include <hip/hip_bf16.h>

typedef __attribute__((ext_vector_type(2))) float v2f;
typedef __attribute__((ext_vector_type(8))) float v8f;

#define H      256
#define NF     9
#define NVOCAB 119
#define NGRAPH 256
#define NLAYER 4
#define BN_EPS 1e-5f

// ---------------------------------------------------------------------------
// Utility: zero a float buffer
// ---------------------------------------------------------------------------
__global__ void zero_kernel(float* __restrict__ p, int n) {
    int i = blockIdx.x * blockDim.x + threadIdx.x;
    if (i < n) p[i] = 0.0f;
}

// ---------------------------------------------------------------------------
// AtomEncoder: h[n,c] = sum_f emb[f, x[n,f], c]
// ---------------------------------------------------------------------------
__global__ void encode_kernel(const int* __restrict__ x,
                              const float* __restrict__ emb,
                              float* __restrict__ h, int N) {
    int n = blockIdx.x;
    int c = threadIdx.x;
    if (n >= N) return;
    float s = 0.0f;
#pragma unroll
    for (int f = 0; f < NF; ++f) {
        int idx = x[n * NF + f];
        s += emb[((long)f * NVOCAB + idx) * H + c];
    }
    h[n * H + c] = s;
}

// ---------------------------------------------------------------------------
// Degree / GCN normalization: nrm = rsqrt(1 + in_degree)
// ---------------------------------------------------------------------------
__global__ void deg_init_kernel(float* __restrict__ nrm, int N) {
    int i = blockIdx.x * blockDim.x + threadIdx.x;
    if (i < N) nrm[i] = 1.0f;   // self-loop
}
__global__ void deg_add_kernel(const int* __restrict__ ei, float* __restrict__ nrm, int E) {
    int e = blockIdx.x * blockDim.x + threadIdx.x;
    if (e < E) atomicAdd(&nrm[ei[E + e]], 1.0f);   // dst row of edge_index
}
__global__ void deg_fin_kernel(float* __restrict__ nrm, int N) {
    int i = blockIdx.x * blockDim.x + threadIdx.x;
    if (i < N) nrm[i] = rsqrtf(nrm[i]);
}

// ---------------------------------------------------------------------------
// Dense GEMM  C[N,256] = A[N,256] @ B[256,256]  using f32 WMMA 16x16x4.
// Block = 512 threads = 16 waves. Block handles a 16-row strip of A (staged
// in LDS once), wave w computes the 16x16 tile at columns [16w,16w+16).
// K loop: 256 / 4 = 64 WMMA ops per wave, unrolled 8x for long clauses.
// Row-range checks are wave-uniform and hoisted to one scalar branch so the
// hot path is a straight load/WMMA/store stream (no per-element exec churn).
// VGPR layouts per CDNA5 ISA 7.12.2:
//   A 16x4 : lanes 0-15 -> M=lane, v0=K0,v1=K1 ; lanes 16-31 -> v0=K2,v1=K3
//   B 4x16 : lane -> N,  v0=K0(half0)/K2(half1), v1=K1/K3
//   C 16x16: VGPR v, lanes 0-15 -> M=v ; lanes 16-31 -> M=v+8 ; N=lane%16
// ---------------------------------------------------------------------------
__global__ __launch_bounds__(512)
void gemm_wmma_kernel(const float* __restrict__ A,
                      const float* __restrict__ B,
                      float* __restrict__ C, int N) {
    __shared__ float As[16][H + 4];   // +4 floats pad: row stride 1040B (16B aligned)

    const int  tid      = threadIdx.x;
    const int  row0     = blockIdx.x * 16;
    const bool fullTile = (row0 + 16 <= N);   // wave-uniform

    // Cooperative load of the 16 x 256 A strip (float4, coalesced).
    if (fullTile) {
        for (int i = tid; i < 16 * (H / 4); i += 512) {
            int r  = i >> 6;            // 64 float4 per row
            int c4 = (i & 63) << 2;
            *(float4*)(&As[r][c4]) = *(const float4*)(A + (long)(row0 + r) * H + c4);
        }
    } else {
        for (int i = tid; i < 16 * (H / 4); i += 512) {
            int r  = i >> 6;
            int c4 = (i & 63) << 2;
            int gr = row0 + r;
            float4 v = make_float4(0.f, 0.f, 0.f, 0.f);
            if (gr < N) v = *(const float4*)(A + (long)gr * H + c4);
            *(float4*)(&As[r][c4]) = v;
        }
    }
    __syncthreads();

    const int wave = tid >> 5;        // 0..15 -> 16-column tile
    const int lane = tid & 31;
    const int half = lane >> 4;       // 0 or 1
    const int lid  = lane & 15;
    const int col  = (wave << 4) + lid;

    v8f acc = {};
#pragma unroll 8
    for (int k0 = 0; k0 < H; k0 += 4) {
        const int ka = k0 + 2 * half;
        v2f a, bb;
        a[0]  = As[lid][ka];
        a[1]  = As[lid][ka + 1];
        bb[0] = B[(long)ka * H + col];
        bb[1] = B[(long)(ka + 1) * H + col];
        acc = __builtin_amdgcn_wmma_f32_16x16x4_f32(
            /*neg_a=*/false, a, /*neg_b=*/false, bb,
            /*c_mod=*/(short)0, acc, /*reuse_a=*/false, /*reuse_b=*/false);
    }

    float* __restrict__ Cp = C + (long)(row0 + (half << 3)) * H + col;
    if (fullTile) {
#pragma unroll
        for (int v = 0; v < 8; ++v) Cp[(long)v * H] = acc[v];
    } else {
#pragma unroll
        for (int v = 0; v < 8; ++v) {
            int r = row0 + (half << 3) + v;
            if (r < N) C[(long)r * H + col] = acc[v];
        }
    }
}

// ---------------------------------------------------------------------------
// agg = hw * self_w + bias   (self_w = nrm^2)
// ---------------------------------------------------------------------------
__global__ void agg_init_kernel(const float* __restrict__ hw,
                                const float* __restrict__ nrm,
                                const float* __restrict__ bias,
                                float* __restrict__ agg, int N) {
    int n = blockIdx.x;
    int c = threadIdx.x;
    if (n >= N) return;
    float sw = nrm[n] * nrm[n];
    agg[n * H + c] = hw[n * H + c] * sw + bias[c];
}

// ---------------------------------------------------------------------------
// Edge scatter: agg[dst] += hw[src] * (nrm[src]*nrm[dst]); 1 block / edge,
// 256-channel coalesced row gather + coalesced float atomics (L2 resident).
// ---------------------------------------------------------------------------
__global__ void agg_edge_kernel(const float* __restrict__ hw,
                                const float* __restrict__ nrm,
                                const int* __restrict__ ei,
                                float* __restrict__ agg, int E) {
    int e = blockIdx.x;
    if (e >= E) return;
    int s = ei[e];
    int d = ei[E + e];
    float w = nrm[s] * nrm[d];
    int c = threadIdx.x;
    atomicAdd(&agg[(long)d * H + c], hw[(long)s * H + c] * w);
}

// ---------------------------------------------------------------------------
// BatchNorm stats: per-channel sum and sum-of-squares (coalesced row chunks,
// one atomic per channel per block).
// ---------------------------------------------------------------------------
__global__ void bn_stats_kernel(const float* __restrict__ agg,
                                float* __restrict__ stats, int N) {
    int c  = threadIdx.x;
    int n0 = blockIdx.x * 32;
    float s1 = 0.0f, s2 = 0.0f;
    for (int r = 0; r < 32; ++r) {
        int n = n0 + r;
        if (n < N) {
            float v = agg[(long)n * H + c];
            s1 += v;
            s2 += v * v;
        }
    }
    atomicAdd(&stats[c], s1);
    atomicAdd(&stats[H + c], s2);
}

// ---------------------------------------------------------------------------
// Fused BN -> ReLU -> residual:  h = relu((agg-mu)*rsqrt(var+eps)*g + b) + h
// ---------------------------------------------------------------------------
__global__ void bn_apply_kernel(const float* __restrict__ agg,
                                const float* __restrict__ stats,
                                const float* __restrict__ gamma,
                                const float* __restrict__ beta,
                                float* __restrict__ h, int N, float invN) {
    int n = blockIdx.x;
    int c = threadIdx.x;
    if (n >= N) return;
    float mu  = stats[c] * invN;
    float var = stats[H + c] * invN - mu * mu;
    float v = (agg[(long)n * H + c] - mu) * rsqrtf(var + BN_EPS) * gamma[c] + beta[c];
    h[(long)n * H + c] = fmaxf(v, 0.0f) + h[(long)n * H + c];
}

// ---------------------------------------------------------------------------
// Global mean pool (atomic accumulate).
// ---------------------------------------------------------------------------
__global__ void pool_kernel(const float* __restrict__ h,
                            const int* __restrict__ batch,
                            float* __restrict__ psum,
                            float* __restrict__ pcnt, int N) {
    int n = blockIdx.x;
    int c = threadIdx.x;
    if (n >= N) return;
    int g = batch[n];
    atomicAdd(&psum[(long)g * H + c], h[(long)n * H + c]);
    if (c == 0) atomicAdd(&pcnt[g], 1.0f);
}

// ---------------------------------------------------------------------------
// MLP head
// ---------------------------------------------------------------------------
__global__ void mlp1_kernel(const float* __restrict__ psum,
                            const float* __restrict__ pcnt,
                            const float* __restrict__ W1,
                            const float* __restrict__ b1,
                            float* __restrict__ out1) {
    __shared__ float gs[H];
    int gi = blockIdx.x;
    float cnt = fmaxf(pcnt[gi], 1.0f);
    for (int i = threadIdx.x; i < H; i += blockDim.x)
        gs[i] = psum[(long)gi * H + i] / cnt;
    __syncthreads();
    int j = threadIdx.x;                 // 128 threads
    float s = b1[j];
    for (int k = 0; k < H; ++k) s += gs[k] * W1[k * 128 + j];
    out1[gi * 128 + j] = fmaxf(s, 0.0f);
}

__global__ void mlp2_kernel(const float* __restrict__ in1,
                            const float* __restrict__ W2,
                            const float* __restrict__ b2,
                            float* __restrict__ out2) {
    __shared__ float gs[128];
    int gi = blockIdx.x;
    for (int i = threadIdx.x; i < 128; i += blockDim.x)
        gs[i] = in1[gi * 128 + i];
    __syncthreads();
    int j = threadIdx.x;                 // 64 threads
    float s = b2[j];
    for (int k = 0; k < 128; ++k) s += gs[k] * W2[k * 64 + j];
    out2[gi * 64 + j] = fmaxf(s, 0.0f);
}

__global__ void mlp3_kernel(const float* __restrict__ in2,
                            const float* __restrict__ W3,
                            const float* __restrict__ b3,
                            float* __restrict__ out) {
    __shared__ float red[64];
    int gi = blockIdx.x;
    int t  = threadIdx.x;                // 64 threads
    red[t] = in2[gi * 64 + t] * W3[t];
    __syncthreads();
    for (int s = 32; s > 0; s >>= 1) {
        if (t < s) red[t] += red[t + s];
        __syncthreads();
    }
    if (t == 0) out[gi] = red[0] + b3[0];
}

// ---------------------------------------------------------------------------
// Launcher
// ---------------------------------------------------------------------------
extern "C" void kernel_launch(void* const* d_in, const int* in_sizes, int n_in,
                              void* d_out, int out_size, void* d_ws, size_t ws_size,
                              hipStream_t stream) {
    const int*   x     = (const int*)d_in[0];
    const int*   ei    = (const int*)d_in[1];
    const int*   batch = (const int*)d_in[2];
    const float* emb   = (const float*)d_in[3];
    const float* W     = (const float*)d_in[4];
    const float* b     = (const float*)d_in[5];
    const float* gamma = (const float*)d_in[6];
    const float* beta  = (const float*)d_in[7];
    const float* W1    = (const float*)d_in[8];
    const float* b1    = (const float*)d_in[9];
    const float* W2    = (const float*)d_in[10];
    const float* b2    = (const float*)d_in[11];
    const float* W3    = (const float*)d_in[12];
    const float* b3    = (const float*)d_in[13];
    float*       out   = (float*)d_out;

    const int N = in_sizes[0] / NF;
    const int E = in_sizes[1] / 2;

    // Workspace carve (floats)
    float* ws    = (float*)d_ws;
    float* h     = ws;                        // N*H
    float* hw    = h    + (long)N * H;        // N*H
    float* agg   = hw   + (long)N * H;        // N*H
    float* nrm   = agg  + (long)N * H;        // N
    float* stats = nrm  + N;                  // 2*H
    float* psum  = stats + 2 * H;             // NGRAPH*H
    float* pcnt  = psum + (long)NGRAPH * H;   // NGRAPH
    float* g1    = pcnt + NGRAPH;             // NGRAPH*128
    float* g2    = g1   + NGRAPH * 128;       // NGRAPH*64

    const int rowTiles = (N + 15) / 16;

    // Encoder + degree normalization
    encode_kernel<<<N, H, 0, stream>>>(x, emb, h, N);
    deg_init_kernel<<<(N + 255) / 256, 256, 0, stream>>>(nrm, N);
    deg_add_kernel<<<(E + 255) / 256, 256, 0, stream>>>(ei, nrm, E);
    deg_fin_kernel<<<(N + 255) / 256, 256, 0, stream>>>(nrm, N);

    // GCN layers
    for (int l = 0; l < NLAYER; ++l) {
        gemm_wmma_kernel<<<rowTiles, 512, 0, stream>>>(h, W + (long)l * H * H, hw, N);
        agg_init_kernel<<<N, H, 0, stream>>>(hw, nrm, b + l * H, agg, N);
        agg_edge_kernel<<<E, H, 0, stream>>>(hw, nrm, ei, agg, E);
        zero_kernel<<<(2 * H + 255) / 256, 256, 0, stream>>>(stats, 2 * H);
        bn_stats_kernel<<<(N + 31) / 32, H, 0, stream>>>(agg, stats, N);
        bn_apply_kernel<<<N, H, 0, stream>>>(agg, stats, gamma + l * H, beta + l * H,
                                             h, N, 1.0f / (float)N);
    }

    // Mean pool + MLP head
    zero_kernel<<<(NGRAPH * H + NGRAPH + 255) / 256, 256, 0, stream>>>(psum, NGRAPH * H + NGRAPH);
    pool_kernel<<<N, H, 0, stream>>>(h, batch, psum, pcnt, N);
    mlp1_kernel<<<NGRAPH, 128, 0, stream>>>(psum, pcnt, W1, b1, g1);
    mlp2_kernel<<<NGRAPH, 64, 0, stream>>>(g1, W2, b2, g2);
    mlp3_kernel<<<NGRAPH, 64, 0, stream>>>(g2, W3, b3, out);
}